// GCN_2473901162893
// MI455X (gfx1250) — compile-verified
//
#include <hip/hip_runtime.h>

typedef __attribute__((ext_vector_type(2))) float v2f;
typedef __attribute__((ext_vector_type(8))) float v8f;

// ---------------------------------------------------------------------------
// Utility: fill a float buffer with a constant
// ---------------------------------------------------------------------------
__global__ __launch_bounds__(256) void k_fill(float* __restrict__ p, float v, long n) {
  long i = (long)blockIdx.x * blockDim.x + threadIdx.x;
  if (i < n) p[i] = v;
}

// ---------------------------------------------------------------------------
// Degree count: deg[col[e]] += 1  (deg pre-filled with 1.0 for self-loops)
// ---------------------------------------------------------------------------
__global__ __launch_bounds__(256) void k_deg_count(const long long* __restrict__ col,
                                                   float* __restrict__ deg, int E) {
  int e = blockIdx.x * blockDim.x + threadIdx.x;
  if (e < E) atomicAdd(&deg[col[e]], 1.0f);
}

__global__ __launch_bounds__(256) void k_rsqrt(float* __restrict__ d, int n) {
  int i = blockIdx.x * blockDim.x + threadIdx.x;
  if (i < n) d[i] = rsqrtf(d[i]);
}

// ---------------------------------------------------------------------------
// Dense GEMM  Out[nrows,128] = X[nrows,128] @ W[128,128]  via fp32 WMMA.
// One wave -> 16 rows x 128 cols (8 accumulator tiles of 16x16).
// W is staged in LDS (64 KB). K=128 as 32 steps of V_WMMA_F32_16X16X4_F32.
//
// ISA fragment layouts (32-bit, wave32):
//   A (16x4):  lane<16: {K0,K1} of row M=lane;  lane>=16: {K2,K3} of row M=lane-16
//   B (4x16):  VGPR0: K0 (lanes0-15)|K2 (lanes16-31); VGPR1: K1|K3, col N=lane&15
//   C/D (16x16): VGPR r: M=r (lanes0-15), M=r+8 (lanes16-31), N=lane&15
// ---------------------------------------------------------------------------
__global__ __launch_bounds__(256) void k_gemm128_wmma(const float* __restrict__ X,
                                                      const float* __restrict__ W,
                                                      float* __restrict__ Out,
                                                      int nrows) {
  __shared__ float sW[128 * 128];
  for (int i = threadIdx.x; i < 128 * 128; i += 256) sW[i] = W[i];
  __syncthreads();

  const int lane = threadIdx.x & 31;
  const int wave = threadIdx.x >> 5;
  const int rowTile = (blockIdx.x * 8 + wave) * 16;
  if (rowTile >= nrows) return;

  const int m    = lane & 15;      // row (A) / col (B,C,D) handled by this lane
  const int hi   = lane >> 4;      // half-wave select
  const int kOff = hi * 2;         // K offset pattern {0,1} vs {2,3}

  v8f acc[8] = {};                 // 8 tiles of 16x16 -> full 128-col stripe

  const float* xrow = X + (size_t)(rowTile + m) * 128;

  for (int k = 0; k < 128; k += 4) {
    const v2f a = *(const v2f*)(xrow + k + kOff);   // A fragment (b64 load)
#pragma unroll
    for (int t = 0; t < 8; ++t) {
      v2f b;
      const int n = t * 16 + m;
      b.x = sW[(k + kOff) * 128 + n];               // B fragment from LDS
      b.y = sW[(k + kOff + 1) * 128 + n];
      acc[t] = __builtin_amdgcn_wmma_f32_16x16x4_f32(
          /*neg_a=*/false, a, /*neg_b=*/false, b,
          /*c_mod=*/(short)0, acc[t], /*reuse_a=*/false, /*reuse_b=*/false);
    }
  }

#pragma unroll
  for (int t = 0; t < 8; ++t) {
#pragma unroll
    for (int r = 0; r < 8; ++r) {
      const int ml = r + hi * 8;
      Out[(size_t)(rowTile + ml) * 128 + t * 16 + m] = acc[t][r];
    }
  }
}

// ---------------------------------------------------------------------------
// Self-loop init: agg[i, f] = h[i, f] * dinv[i]^2
// ---------------------------------------------------------------------------
__global__ __launch_bounds__(256) void k_selfloop(const float* __restrict__ h,
                                                  const float* __restrict__ dinv,
                                                  float* __restrict__ agg, long total) {
  long i = (long)blockIdx.x * blockDim.x + threadIdx.x;
  if (i < total) {
    const float s = dinv[i >> 7];
    agg[i] = h[i] * s * s;
  }
}

// ---------------------------------------------------------------------------
// Edge scatter: agg[col] += h[row] * dinv[row] * dinv[col]
// One wave per edge; each lane moves a float4 (128 features / 32 lanes).
// ---------------------------------------------------------------------------
__global__ __launch_bounds__(256) void k_scatter(const float* __restrict__ h,
                                                 const long long* __restrict__ row,
                                                 const long long* __restrict__ col,
                                                 const float* __restrict__ dinv,
                                                 float* __restrict__ agg, int E) {
  const long gid = (long)blockIdx.x * blockDim.x + threadIdx.x;
  const long e = gid >> 5;
  const int lane = (int)(gid & 31);
  if (e >= E) return;
  const long long r = row[e];
  const long long c = col[e];
  const float w = dinv[r] * dinv[c];
  const float4 v = ((const float4*)(h + (size_t)r * 128))[lane];
  float* dst = agg + (size_t)c * 128 + lane * 4;
  atomicAdd(dst + 0, v.x * w);
  atomicAdd(dst + 1, v.y * w);
  atomicAdd(dst + 2, v.z * w);
  atomicAdd(dst + 3, v.w * w);
}

// ---------------------------------------------------------------------------
// Bias + ReLU in place
// ---------------------------------------------------------------------------
__global__ __launch_bounds__(256) void k_bias_relu(float* __restrict__ h,
                                                   const float* __restrict__ b, long total) {
  long i = (long)blockIdx.x * blockDim.x + threadIdx.x;
  if (i < total) h[i] = fmaxf(h[i] + b[i & 127], 0.0f);
}

// ---------------------------------------------------------------------------
// Global max pool: pool[batch[i]] = max(pool[batch[i]], h[i]) per feature.
// h >= 0 (post-ReLU) so uint bit-pattern atomicMax is exact IEEE max.
// One wave per node, float4 per lane.
// ---------------------------------------------------------------------------
__global__ __launch_bounds__(256) void k_pool_max(const float* __restrict__ h,
                                                  const long long* __restrict__ batch,
                                                  float* __restrict__ pool, int n) {
  const long gid = (long)blockIdx.x * blockDim.x + threadIdx.x;
  const long node = gid >> 5;
  const int lane = (int)(gid & 31);
  if (node >= n) return;
  const long long g = batch[node];
  const float4 v = ((const float4*)(h + (size_t)node * 128))[lane];
  unsigned* dst = (unsigned*)(pool + (size_t)g * 128 + lane * 4);
  atomicMax(dst + 0, __float_as_uint(v.x));
  atomicMax(dst + 1, __float_as_uint(v.y));
  atomicMax(dst + 2, __float_as_uint(v.z));
  atomicMax(dst + 3, __float_as_uint(v.w));
}

// ---------------------------------------------------------------------------
// Head: out[g,c] = pool[g,:] . Wlin[:,c] + blin[c]   ([512,128]@[128,10])
// ---------------------------------------------------------------------------
__global__ __launch_bounds__(256) void k_head(const float* __restrict__ pool,
                                              const float* __restrict__ Wlin,
                                              const float* __restrict__ blin,
                                              float* __restrict__ out, int G, int C) {
  const int t = blockIdx.x * blockDim.x + threadIdx.x;
  if (t >= G * C) return;
  const int g = t / C;
  const int c = t % C;
  const float* p = pool + (size_t)g * 128;
  float s = blin[c];
#pragma unroll 8
  for (int k = 0; k < 128; ++k) s += p[k] * Wlin[k * C + c];
  out[t] = s;
}

// ---------------------------------------------------------------------------
extern "C" void kernel_launch(void* const* d_in, const int* in_sizes, int n_in,
                              void* d_out, int out_size, void* d_ws, size_t ws_size,
                              hipStream_t stream) {
  const float*      x     = (const float*)d_in[0];
  const long long*  ei    = (const long long*)d_in[1];
  const long long*  batch = (const long long*)d_in[2];
  const float*      W1    = (const float*)d_in[3];
  const float*      b1    = (const float*)d_in[4];
  const float*      W2    = (const float*)d_in[5];
  const float*      b2    = (const float*)d_in[6];
  const float*      Wlin  = (const float*)d_in[7];
  const float*      blin  = (const float*)d_in[8];

  const int N = in_sizes[0] / 128;
  const int E = in_sizes[1] / 2;
  const int C = in_sizes[8];
  const int G = out_size / C;

  const long long* row = ei;       // edge_index[0]
  const long long* col = ei + E;   // edge_index[1]

  // Workspace layout (floats): dinv[N] | bufA[N*128] | bufB[N*128] | pool[G*128]
  float* dinv = (float*)d_ws;
  float* bufA = dinv + (((size_t)N + 63) & ~(size_t)63);
  float* bufB = bufA + (size_t)N * 128;
  float* pool = bufB + (size_t)N * 128;

  const long NF = (long)N * 128;
  const long PF = (long)G * 128;

  // ---- normalization: dinv = rsqrt(1 + indegree) ----
  k_fill<<<(N + 255) / 256, 256, 0, stream>>>(dinv, 1.0f, N);
  k_deg_count<<<(E + 255) / 256, 256, 0, stream>>>(col, dinv, E);
  k_rsqrt<<<(N + 255) / 256, 256, 0, stream>>>(dinv, N);

  const int gemmBlocks = (N + 127) / 128;          // 8 waves x 16 rows per block
  const long elemBlocks = (NF + 255) / 256;
  const long edgeBlocks = ((long)E * 32 + 255) / 256;

  // ---- layer 1: h1 = relu(D^-1/2 A D^-1/2 (x@W1) + b1) ----
  k_gemm128_wmma<<<gemmBlocks, 256, 0, stream>>>(x, W1, bufA, N);
  k_selfloop<<<elemBlocks, 256, 0, stream>>>(bufA, dinv, bufB, NF);
  k_scatter<<<edgeBlocks, 256, 0, stream>>>(bufA, row, col, dinv, bufB, E);
  k_bias_relu<<<elemBlocks, 256, 0, stream>>>(bufB, b1, NF);

  // ---- layer 2 ----
  k_gemm128_wmma<<<gemmBlocks, 256, 0, stream>>>(bufB, W2, bufA, N);
  k_selfloop<<<elemBlocks, 256, 0, stream>>>(bufA, dinv, bufB, NF);
  k_scatter<<<edgeBlocks, 256, 0, stream>>>(bufA, row, col, dinv, bufB, E);
  k_bias_relu<<<elemBlocks, 256, 0, stream>>>(bufB, b2, NF);

  // ---- global max pool + head ----
  k_fill<<<(PF + 255) / 256, 256, 0, stream>>>(pool, 0.0f, PF);
  k_pool_max<<<((long)N * 32 + 255) / 256, 256, 0, stream>>>(bufB, batch, pool, N);
  k_head<<<(G * C + 255) / 256, 256, 0, stream>>>(pool, Wlin, blin, (float*)d_out, G, C);
}